// SampleAndAggregate_65068754534945
// MI455X (gfx1250) — compile-verified
//
#include <hip/hip_runtime.h>

// GraphSAGE sample+aggregate for MI455X (gfx1250, wave32).
// f32 end-to-end; GEMMs run on V_WMMA_F32_16X16X4_F32 with K-pair-interleaved
// LDS weight staging so every B fragment is one aligned ds_load_b64.

#define NFEAT 128
#define DEG   128
#define BATCH 1024
#define SAMP1 10
#define SAMP2 25
#define NH1   (BATCH * SAMP1)      // 10240
#define NH2   (NH1 * SAMP2)        // 256000
#define OUTLD 256

typedef __attribute__((ext_vector_type(2))) float v2f;
typedef __attribute__((ext_vector_type(8))) float v8f;

// ---------- sampling: out[i] = adj[parent[i/S]*DEG + col[i]] ----------
__global__ void sample_kernel(const int* __restrict__ adj,
                              const int* __restrict__ parent,
                              const int* __restrict__ col,
                              int* __restrict__ out, int S, int total) {
    int i = blockIdx.x * blockDim.x + threadIdx.x;
    if (i >= total) return;
    int p = parent[i / S];
    out[i] = adj[(size_t)p * DEG + col[i]];
}

// ---------- gather rows: 4 rows/block, 32 lanes x float4 = one 512B row ----------
__global__ void gather_rows_kernel(const float4* __restrict__ feat,
                                   const int* __restrict__ idx,
                                   float4* __restrict__ out) {
    int r = blockIdx.x * 4 + threadIdx.y;
    int lane = threadIdx.x;                       // 0..31
    out[(size_t)r * 32 + lane] = feat[(size_t)idx[r] * 32 + lane];
}

// ---------- fused gather + mean over S neighbors per output row ----------
__global__ void gather_mean_kernel(const float4* __restrict__ feat,
                                   const int* __restrict__ idx,
                                   float4* __restrict__ out, int S, float inv) {
    int r = blockIdx.x * 4 + threadIdx.y;
    int lane = threadIdx.x;                       // 0..31
    float ax = 0.f, ay = 0.f, az = 0.f, aw = 0.f;
    for (int j = 0; j < S; ++j) {
        int nid = idx[r * S + j];
        float4 v = feat[(size_t)nid * 32 + lane];
        ax += v.x; ay += v.y; az += v.z; aw += v.w;
    }
    out[(size_t)r * 32 + lane] = make_float4(ax * inv, ay * inv, az * inv, aw * inv);
}

// ---------- mean over SAMP1 consecutive 256-wide rows ----------
__global__ void mean_rows256_kernel(const float4* __restrict__ in,
                                    float4* __restrict__ out, float inv) {
    int b = blockIdx.x * 4 + threadIdx.y;
    int t = threadIdx.x;                          // 0..63  (64*float4 = 256 f32)
    float ax = 0.f, ay = 0.f, az = 0.f, aw = 0.f;
    for (int j = 0; j < SAMP1; ++j) {
        float4 v = in[((size_t)(b * SAMP1 + j)) * 64 + t];
        ax += v.x; ay += v.y; az += v.z; aw += v.w;
    }
    out[(size_t)b * 64 + t] = make_float4(ax * inv, ay * inv, az * inv, aw * inv);
}

// ---------- WMMA f32 GEMM: Out[:, col_off:col_off+128] = act(X[rows,K] @ W[K,128])
// block = 128 threads (4 waves); each wave: 16 rows x 128 cols (8 N-tiles).
// LDS holds one 128-deep K chunk of W, K-pair interleaved:
//   ldsW[pr*PSTRIDE + 2c] = W[k0+2pr][c], ldsW[pr*PSTRIDE + 2c+1] = W[k0+2pr+1][c]
// PSTRIDE=288 (mod 64 == 32) -> the two half-waves (pair rows pr, pr+1) hit
// disjoint 32-bank halves: conflict-free ds_load_b64 B fragments.
#define PSTRIDE 288
__global__ void wmma_gemm_kernel(const float* __restrict__ X,
                                 const float* __restrict__ W,
                                 float* __restrict__ Out,
                                 int K, int col_off, int do_relu) {
    __shared__ float ldsW[64 * PSTRIDE];          // 72 KB of the 320 KB/WGP
    const int tid   = threadIdx.x;
    const int lane  = tid & 31;
    const int wave  = tid >> 5;
    const int row0  = (blockIdx.x * 4 + wave) * 16;
    const int m     = lane & 15;                  // A row / B col within tile
    const int half  = lane >> 4;                  // 0 or 1 (half-wave)
    const int khalf = half * 2;                   // A-frag K split (K, K+1 | K+2, K+3)
    const float* __restrict__ xrow  = X + (size_t)(row0 + m) * K;
    const float* __restrict__ bbase = ldsW + half * PSTRIDE + 2 * m;

    v8f acc[8] = {};

    for (int k0 = 0; k0 < K; k0 += 128) {
        if (k0 + 128 < K) __builtin_prefetch(xrow + k0 + 128, 0, 3);

        // stage + interleave W[k0..k0+127][0..127]: 64 pair-rows x 32 col-quads
        for (int u = tid; u < 64 * 32; u += 128) {
            int r  = u >> 5;                      // pair row (k = k0+2r, k0+2r+1)
            int cq = (u & 31) << 2;               // col quad start
            float4 a = *(const float4*)(W + (size_t)(k0 + 2 * r) * 128 + cq);
            float4 b = *(const float4*)(W + (size_t)(k0 + 2 * r + 1) * 128 + cq);
            float* d = ldsW + r * PSTRIDE + 2 * cq;
            *(float4*)(d)     = make_float4(a.x, b.x, a.y, b.y);
            *(float4*)(d + 4) = make_float4(a.z, b.z, a.w, b.w);
        }
        __syncthreads();

        for (int kk = 0; kk < 128; kk += 4) {
            // A fragment: X[row, kk+khalf], X[row, kk+khalf+1] -> one global b64
            v2f a = *(const v2f*)(xrow + k0 + kk + khalf);
            const float* bp = bbase + (kk >> 1) * PSTRIDE;
#pragma unroll
            for (int n = 0; n < 8; ++n) {
                // B fragment: (W[kk+khalf][n*16+m], W[kk+khalf+1][n*16+m])
                v2f b = *(const v2f*)(bp + n * 32);   // one ds_load_b64
                acc[n] = __builtin_amdgcn_wmma_f32_16x16x4_f32(
                    false, a, false, b, (short)0, acc[n], false, false);
            }
        }
        __syncthreads();
    }

    // C/D layout: vgpr j = rows row0+j (lanes 0-15) / row0+j+8 (lanes 16-31)
    const int rbase = row0 + half * 8;
#pragma unroll
    for (int n = 0; n < 8; ++n) {
        int ocol = col_off + n * 16 + m;
#pragma unroll
        for (int j = 0; j < 8; ++j) {
            float v = acc[n][j];
            if (do_relu) v = fmaxf(v, 0.f);
            Out[(size_t)(rbase + j) * OUTLD + ocol] = v;
        }
    }
}

extern "C" void kernel_launch(void* const* d_in, const int* in_sizes, int n_in,
                              void* d_out, int out_size, void* d_ws, size_t ws_size,
                              hipStream_t stream) {
    (void)in_sizes; (void)n_in; (void)out_size; (void)ws_size;
    const float* features = (const float*)d_in[0];
    const int*   adj      = (const int*)d_in[1];
    const int*   batch    = (const int*)d_in[2];
    const int*   col1     = (const int*)d_in[3];
    const int*   col2     = (const int*)d_in[4];
    const float* w_neigh0 = (const float*)d_in[5];
    const float* w_self0  = (const float*)d_in[6];
    const float* w_neigh1 = (const float*)d_in[7];
    const float* w_self1  = (const float*)d_in[8];
    float* out = (float*)d_out;

    // workspace layout (all 16B aligned)
    char* p = (char*)d_ws;
    int*   s1    = (int*)p;   p += (size_t)NH1 * 4;                 // 40 KB
    int*   s2    = (int*)p;   p += (size_t)NH2 * 4;                 // 1.0 MB
    float* h0    = (float*)p; p += (size_t)BATCH * NFEAT * 4;       // 0.5 MB
    float* h1    = (float*)p; p += (size_t)NH1   * NFEAT * 4;       // 5.2 MB
    float* mean1 = (float*)p; p += (size_t)BATCH * NFEAT * 4;       // 0.5 MB
    float* mean2 = (float*)p; p += (size_t)NH1   * NFEAT * 4;       // 5.2 MB
    float* h0c   = (float*)p; p += (size_t)BATCH * OUTLD * 4;       // 1.0 MB
    float* h1c   = (float*)p; p += (size_t)NH1   * OUTLD * 4;       // 10.5 MB
    float* m1c   = (float*)p; p += (size_t)BATCH * OUTLD * 4;       // 1.0 MB

    // ---- sampling ----
    sample_kernel<<<(NH1 + 255) / 256, 256, 0, stream>>>(adj, batch, col1, s1, SAMP1, NH1);
    sample_kernel<<<(NH2 + 255) / 256, 256, 0, stream>>>(adj, s1,    col2, s2, SAMP2, NH2);

    // ---- gathers / neighbor means (bandwidth-bound: ~131 MB for hop2) ----
    dim3 gblk(32, 4);
    gather_rows_kernel<<<BATCH / 4, gblk, 0, stream>>>((const float4*)features, batch, (float4*)h0);
    gather_rows_kernel<<<NH1   / 4, gblk, 0, stream>>>((const float4*)features, s1,    (float4*)h1);
    gather_mean_kernel<<<BATCH / 4, gblk, 0, stream>>>((const float4*)features, s1, (float4*)mean1,
                                                       SAMP1, 1.f / SAMP1);
    gather_mean_kernel<<<NH1   / 4, gblk, 0, stream>>>((const float4*)features, s2, (float4*)mean2,
                                                       SAMP2, 1.f / SAMP2);

    // ---- layer 0 (ReLU), shared aggregator weights ----
    wmma_gemm_kernel<<<BATCH / 64, 128, 0, stream>>>(h0,    w_self0,  h0c, NFEAT,   0, 1);
    wmma_gemm_kernel<<<BATCH / 64, 128, 0, stream>>>(mean1, w_neigh0, h0c, NFEAT, 128, 1);
    wmma_gemm_kernel<<<NH1   / 64, 128, 0, stream>>>(h1,    w_self0,  h1c, NFEAT,   0, 1);
    wmma_gemm_kernel<<<NH1   / 64, 128, 0, stream>>>(mean2, w_neigh0, h1c, NFEAT, 128, 1);

    // ---- mean over hop1 of layer-0 hidden ----
    dim3 mblk(64, 4);
    mean_rows256_kernel<<<BATCH / 4, mblk, 0, stream>>>((const float4*)h1c, (float4*)m1c,
                                                        1.f / SAMP1);

    // ---- layer 1 (identity) ----
    wmma_gemm_kernel<<<BATCH / 64, 128, 0, stream>>>(h0c, w_self1,  out, 2 * NFEAT,   0, 0);
    wmma_gemm_kernel<<<BATCH / 64, 128, 0, stream>>>(m1c, w_neigh1, out, 2 * NFEAT, 128, 0);
}